// HashEmbedder_89799176225629
// MI455X (gfx1250) — compile-verified
//
#include <hip/hip_runtime.h>
#include <stdint.h>

#define BLK 256
#define PTS_PER_BLK 16
#define NLEV 16
#define S_DENSE 6
#define T_HASH 524309ull
#define OUT_W 35
#define TILE_F (PTS_PER_BLK * OUT_W)   // 560 floats = 2240 B per block tile

typedef __attribute__((ext_vector_type(4))) unsigned int tdm_v4u;
typedef __attribute__((ext_vector_type(8))) int          tdm_v8i;
typedef __attribute__((ext_vector_type(4))) int          tdm_v4i;

// Per-level grid resolutions: int(16 * 1.38**i)
__constant__ int c_res[NLEV] = {16,22,30,42,58,80,110,152,210,290,400,553,763,1053,1453,2005};
// entries_size = float32(1.0/(res-1)), computed in double then rounded (matches numpy)
__constant__ float c_es[NLEV] = {
  (float)(1.0/15.0),  (float)(1.0/21.0),  (float)(1.0/29.0),  (float)(1.0/41.0),
  (float)(1.0/57.0),  (float)(1.0/79.0),  (float)(1.0/109.0), (float)(1.0/151.0),
  (float)(1.0/209.0), (float)(1.0/289.0), (float)(1.0/399.0), (float)(1.0/552.0),
  (float)(1.0/762.0), (float)(1.0/1052.0),(float)(1.0/1452.0),(float)(1.0/2004.0)};
// cumulative base offsets into the dense table for levels 0..5
__constant__ unsigned int c_base[S_DENSE] = {0u, 4096u, 14744u, 41744u, 115832u, 310944u};

static __device__ __forceinline__ int iclamp(int v, int lo, int hi) {
  return v < lo ? lo : (v > hi ? hi : v);
}

// Build a 1-D TDM descriptor: tile of `tile_elems` f32 at `gaddr`, LDS offset `ldsoff`,
// within a tensor of `tensor_elems` f32 (OOB elements dropped/zeroed by TDM).
static __device__ __forceinline__ void tdm_1d_f32(uint64_t gaddr, uint32_t ldsoff,
                                                  uint32_t tensor_elems, uint32_t tile_elems,
                                                  bool is_store) {
  tdm_v4u g0;
  g0[0] = 1u;                                            // count=1, is_restore=0
  g0[1] = ldsoff;                                        // lds_addr (bytes)
  g0[2] = (uint32_t)(gaddr & 0xffffffffull);             // global_addr[31:0]
  g0[3] = (uint32_t)((gaddr >> 32) & 0x01ffffffull) | (2u << 30); // addr[56:32], type=2

  tdm_v8i g1;
  g1[0] = (int)(2u << 16);                               // workgroup_mask=0, data_size=4B
  g1[1] = (int)((tensor_elems & 0xffffu) << 16);         // tensor_dim0[15:0] @ bits[63:48]
  g1[2] = (int)(((tensor_elems >> 16) & 0xffffu) | (1u << 16)); // dim0[31:16], tensor_dim1=1
  g1[3] = (int)(tile_elems << 16);                       // tile_dim0 @ bits[127:112]
  g1[4] = 0;                                             // tile_dim1/2 unused
  g1[5] = (int)tensor_elems;                             // tensor_dim0_stride[31:0]
  g1[6] = 0;
  g1[7] = 0;

  tdm_v4i g2 = {0, 0, 0, 0};
  tdm_v4i g3 = {0, 0, 0, 0};
#if defined(__clang_major__) && (__clang_major__ >= 23)
  tdm_v8i g4 = {0, 0, 0, 0, 0, 0, 0, 0};
  if (is_store) __builtin_amdgcn_tensor_store_from_lds(g0, g1, g2, g3, g4, 0);
  else          __builtin_amdgcn_tensor_load_to_lds(g0, g1, g2, g3, g4, 0);
#else
  if (is_store) __builtin_amdgcn_tensor_store_from_lds(g0, g1, g2, g3, 0);
  else          __builtin_amdgcn_tensor_load_to_lds(g0, g1, g2, g3, 0);
#endif
}

__global__ __launch_bounds__(BLK) void hash_embed_kernel(
    const float* __restrict__ xyz, const float* __restrict__ dense,
    const float* __restrict__ hasht, float* __restrict__ out, int npts)
{
  __shared__ float s_xyz[PTS_PER_BLK * 3];
  __shared__ float s_out[TILE_F];

  // ---- Stage this block's 48 xyz floats into LDS via the Tensor Data Mover ----
  if (threadIdx.x < 32) {   // wave 0 issues the DMA
    const uint64_t gaddr = (uint64_t)(uintptr_t)xyz +
                           (uint64_t)blockIdx.x * (PTS_PER_BLK * 3 * sizeof(float));
    const uint32_t ldsoff = (uint32_t)(uintptr_t)(void*)s_xyz;  // low 32 bits = LDS offset
    tdm_1d_f32(gaddr, ldsoff, (uint32_t)npts * 3u, PTS_PER_BLK * 3, /*store=*/false);
    __builtin_prefetch(xyz + (size_t)(blockIdx.x + 1) * (PTS_PER_BLK * 3), 0, 1); // warm next tile
    __builtin_amdgcn_s_wait_tensorcnt(0);
    __asm__ __volatile__("" ::: "memory");   // LDS now holds data the compiler can't see
  }
  __syncthreads();

  // Wave-uniform level split: lanes of one wave cover exactly 2 levels,
  // so the dense-vs-hash branch never diverges inside a wave.
  const int l  = threadIdx.x >> 4;   // level 0..15
  const int lp = threadIdx.x & 15;   // local point 0..15
  const int n  = blockIdx.x * PTS_PER_BLK + lp;

  if (n < npts) {
    const float x0 = s_xyz[lp * 3 + 0];
    const float x1 = s_xyz[lp * 3 + 1];
    const float x2 = s_xyz[lp * 3 + 2];

    const int   r  = c_res[l];
    const float es = c_es[l];
    const float f0 = x0 / es, f1 = x1 / es, f2 = x2 / es;  // fp32 divide, as in reference

    int i0 = iclamp((int)f0, 0, r - 1);
    int i1 = iclamp((int)f1, 0, r - 1);
    int i2 = iclamp((int)f2, 0, r - 1);
    int j0 = min(i0 + 1, r - 1);
    int j1 = min(i1 + 1, r - 1);
    int j2 = min(i2 + 1, r - 1);

    // fractional offsets from corner 0; reference multiplies only dims 0 and 1 (replicated)
    const float o0 = f0 - (float)i0;
    const float o1 = f1 - (float)i1;
    const float w00 = (1.0f - o0) * (1.0f - o1);  // corners 0,1 (b0=0,b1=0)
    const float w01 = (1.0f - o0) * o1;           // corners 2,3
    const float w10 = o0 * (1.0f - o1);           // corners 4,5
    const float w11 = o0 * o1;                    // corners 6,7

    float2 v[8];
    if (l < S_DENSE) {
      const float2* tab = (const float2*)dense;
      const uint32_t base = c_base[l];
      const uint32_t s0 = (uint32_t)r * (uint32_t)r, s1 = (uint32_t)r;
      const uint32_t a0 = (uint32_t)i0 * s0, b0 = (uint32_t)j0 * s0;
      const uint32_t a1 = (uint32_t)i1 * s1, b1 = (uint32_t)j1 * s1;
      const uint32_t a2 = (uint32_t)i2,      b2 = (uint32_t)j2;
      uint32_t idx[8];
      idx[0] = base + a0 + a1 + a2;  idx[1] = base + a0 + a1 + b2;
      idx[2] = base + a0 + b1 + a2;  idx[3] = base + a0 + b1 + b2;
      idx[4] = base + b0 + a1 + a2;  idx[5] = base + b0 + a1 + b2;
      idx[6] = base + b0 + b1 + a2;  idx[7] = base + b0 + b1 + b2;
#pragma unroll
      for (int k = 0; k < 8; ++k) v[k] = tab[idx[k]];    // 8 b64 gathers issued back-to-back
    } else {
      const float2* tab = (const float2*)hasht + (size_t)(l - S_DENSE) * T_HASH;
      // ih = 1 ^ x0*1 ^ x1*19349663 ^ x2*83492791 (int64, products up to 2^38) mod 524309
      const uint64_t h0a = (uint64_t)(uint32_t)i0;
      const uint64_t h0b = (uint64_t)(uint32_t)j0;
      const uint64_t h1a = (uint64_t)(uint32_t)i1 * 19349663ull;
      const uint64_t h1b = (uint64_t)(uint32_t)j1 * 19349663ull;
      const uint64_t h2a = (uint64_t)(uint32_t)i2 * 83492791ull;
      const uint64_t h2b = (uint64_t)(uint32_t)j2 * 83492791ull;
      uint32_t idx[8];
      idx[0] = (uint32_t)((1ull ^ h0a ^ h1a ^ h2a) % T_HASH);
      idx[1] = (uint32_t)((1ull ^ h0a ^ h1a ^ h2b) % T_HASH);
      idx[2] = (uint32_t)((1ull ^ h0a ^ h1b ^ h2a) % T_HASH);
      idx[3] = (uint32_t)((1ull ^ h0a ^ h1b ^ h2b) % T_HASH);
      idx[4] = (uint32_t)((1ull ^ h0b ^ h1a ^ h2a) % T_HASH);
      idx[5] = (uint32_t)((1ull ^ h0b ^ h1a ^ h2b) % T_HASH);
      idx[6] = (uint32_t)((1ull ^ h0b ^ h1b ^ h2a) % T_HASH);
      idx[7] = (uint32_t)((1ull ^ h0b ^ h1b ^ h2b) % T_HASH);
#pragma unroll
      for (int k = 0; k < 8; ++k) v[k] = tab[idx[k]];
    }

    const float accx = w00 * (v[0].x + v[1].x) + w01 * (v[2].x + v[3].x) +
                       w10 * (v[4].x + v[5].x) + w11 * (v[6].x + v[7].x);
    const float accy = w00 * (v[0].y + v[1].y) + w01 * (v[2].y + v[3].y) +
                       w10 * (v[4].y + v[5].y) + w11 * (v[6].y + v[7].y);

    // Stage the block's contiguous 16x35 output tile in LDS (conflict-free: 35 odd).
    s_out[lp * OUT_W + 3 + 2 * l + 0] = accx;
    s_out[lp * OUT_W + 3 + 2 * l + 1] = accy;
    if (l == 0) {                     // include_input: normalized x == x (bounds are [0,1])
      s_out[lp * OUT_W + 0] = x0;
      s_out[lp * OUT_W + 1] = x1;
      s_out[lp * OUT_W + 2] = x2;
    }
  }
  __syncthreads();   // per-wave s_wait_dscnt + barrier: s_out complete before DMA

  // ---- One coalesced 2240 B TDM store of the whole tile: LDS -> global ----
  if (threadIdx.x < 32) {
    const uint32_t tensor_elems = (uint32_t)npts * (uint32_t)OUT_W;
    const uint32_t tile_base    = (uint32_t)blockIdx.x * (uint32_t)TILE_F;
    uint32_t tile_elems = TILE_F;
    if (tile_base + TILE_F > tensor_elems) tile_elems = tensor_elems - tile_base; // partial block
    const uint64_t gaddr = (uint64_t)(uintptr_t)out + (uint64_t)tile_base * sizeof(float);
    const uint32_t ldsoff = (uint32_t)(uintptr_t)(void*)s_out;
    __asm__ __volatile__("" ::: "memory");
    tdm_1d_f32(gaddr, ldsoff, tensor_elems, tile_elems, /*store=*/true);
    __builtin_amdgcn_s_wait_tensorcnt(0);   // S_ENDPGM would also imply wait-idle
  }
}

extern "C" void kernel_launch(void* const* d_in, const int* in_sizes, int n_in,
                              void* d_out, int out_size, void* d_ws, size_t ws_size,
                              hipStream_t stream) {
  const float* xyz   = (const float*)d_in[0];   // (2, 65536, 3) f32
  const float* dense = (const float*)d_in[1];   // (822944, 2) f32
  const float* hasht = (const float*)d_in[2];   // (10, 524309, 2) f32
  float* out = (float*)d_out;                   // (2, 65536, 35) f32
  (void)d_ws; (void)ws_size; (void)n_in; (void)out_size;

  const int npts   = in_sizes[0] / 3;           // 131072
  const int blocks = (npts + PTS_PER_BLK - 1) / PTS_PER_BLK;
  hash_embed_kernel<<<dim3(blocks), dim3(BLK), 0, stream>>>(xyz, dense, hasht, out, npts);
}